// PixelSharp_39118562132479
// MI455X (gfx1250) — compile-verified
//
#include <hip/hip_runtime.h>

// ---------------------------------------------------------------------------
// PixelCNN-style forward for MI455X (gfx1250, wave32).
// All channel-mixing ops go through one WMMA f32 16x16x4 GEMM kernel: each
// wave computes a 16(M) x 64(N) strip = 4 WMMA tiles, reusing the A fragment.
// Addressing is saddr + 32-bit voffset (no per-lane 64-bit pointer math):
// steady state = 1x global_load_b64 (A) + 8x global_load_b32 (B, coalesced)
// -> one wait -> 4x v_wmma -> 4x v_add_nc_u32.
// Conv taps are pre-packed dense so A is always contiguous.
// ---------------------------------------------------------------------------

#define P_TOT 8192          // N*H*W = 8*32*32 spatial columns
#define LDY   8192

typedef float v2f __attribute__((ext_vector_type(2)));
typedef float v8f __attribute__((ext_vector_type(8)));

// GEMM: Y[Co x 8192] (+)= W[Co x Ci] * X(gathered) + bias
// Gather per output column (n,r,cl): pos = n*img_stride + (r+sy)*row_stride + (cl+sx)
// mode: 0 = overwrite (+bias if non-null), 1 = accumulate + bias, 2 = accumulate
__global__ __launch_bounds__(32)
void wmma_gemm(const float* __restrict__ Wm,
               const float* __restrict__ bias,
               const float* __restrict__ X, int chan_stride,
               int img_stride, int row_stride, int sy, int sx,
               float* __restrict__ Y, int Co, int Ci, int mode)
{
    const int lane  = threadIdx.x & 31;
    const int half  = lane >> 4;        // 0: K 0..1 / rows 0..7, 1: K 2..3 / rows 8..15
    const int l15   = lane & 15;
    const int mbase = blockIdx.y << 4;
    const int cbase = blockIdx.x << 6;  // 4 tiles of 16 columns per wave

    auto coloff = [&](int t) {
        int col = cbase + (t << 4) + l15;
        int n   = col >> 10;
        int r   = (col >> 5) & 31;
        int cl  = col & 31;
        return n * img_stride + (r + sy) * row_stride + (cl + sx);
    };

    const int  m     = mbase + l15;
    const bool mok   = (m < Co);
    const int  m_eff = mok ? m : (Co - 1);      // clamp: loads always in-bounds

    const int cs  = chan_stride;
    const int cs4 = cs * 4;
    const float* __restrict__ X0 = X;           // channel kk
    const float* __restrict__ X1 = X + cs;      // channel kk+1
    const v2f*   __restrict__ W2 = (const v2f*)Wm;

    // 32-bit offsets; bases stay uniform (SGPR) -> saddr-form loads.
    int bo0 = coloff(0) + (half << 1) * cs;
    int bo1 = coloff(1) + (half << 1) * cs;
    int bo2 = coloff(2) + (half << 1) * cs;
    int bo3 = coloff(3) + (half << 1) * cs;
    int wo  = (m_eff * Ci + (half << 1)) >> 1;  // v2f index; even whenever Ci even

    v8f acc[4];
#pragma unroll
    for (int t = 0; t < 4; ++t) acc[t] = v8f{0.f,0.f,0.f,0.f,0.f,0.f,0.f,0.f};

    const int kmain = Ci & ~3;
    for (int k = 0; k < kmain; k += 4) {
        v2f a = W2[wo];  wo += 2;
        v2f b0, b1, b2, b3;
        b0[0] = X0[bo0]; b0[1] = X1[bo0];
        b1[0] = X0[bo1]; b1[1] = X1[bo1];
        b2[0] = X0[bo2]; b2[1] = X1[bo2];
        b3[0] = X0[bo3]; b3[1] = X1[bo3];
        bo0 += cs4; bo1 += cs4; bo2 += cs4; bo3 += cs4;
        a[0] = mok ? a[0] : 0.0f;   // v_cndmask, no EXEC games
        a[1] = mok ? a[1] : 0.0f;
        acc[0] = __builtin_amdgcn_wmma_f32_16x16x4_f32(false, a, false, b0, (short)0, acc[0], false, false);
        acc[1] = __builtin_amdgcn_wmma_f32_16x16x4_f32(false, a, false, b1, (short)0, acc[1], false, false);
        acc[2] = __builtin_amdgcn_wmma_f32_16x16x4_f32(false, a, false, b2, (short)0, acc[2], false, false);
        acc[3] = __builtin_amdgcn_wmma_f32_16x16x4_f32(false, a, false, b3, (short)0, acc[3], false, false);
    }

    if (kmain < Ci) {   // tail only for layer-0 GEMMs (Ci = 1,2,3)
        v2f a, b0, b1, b2, b3;
#pragma unroll
        for (int j = 0; j < 2; ++j) {
            int  kk  = kmain + (half << 1) + j;
            bool kok = (kk < Ci);
            int  kc  = kok ? kk : 0;
            float av = Wm[m_eff * Ci + kc];
            a[j]  = (kok && mok) ? av : 0.0f;
            b0[j] = kok ? X[kc * cs + coloff(0)] : 0.0f;
            b1[j] = kok ? X[kc * cs + coloff(1)] : 0.0f;
            b2[j] = kok ? X[kc * cs + coloff(2)] : 0.0f;
            b3[j] = kok ? X[kc * cs + coloff(3)] : 0.0f;
        }
        acc[0] = __builtin_amdgcn_wmma_f32_16x16x4_f32(false, a, false, b0, (short)0, acc[0], false, false);
        acc[1] = __builtin_amdgcn_wmma_f32_16x16x4_f32(false, a, false, b1, (short)0, acc[1], false, false);
        acc[2] = __builtin_amdgcn_wmma_f32_16x16x4_f32(false, a, false, b2, (short)0, acc[2], false, false);
        acc[3] = __builtin_amdgcn_wmma_f32_16x16x4_f32(false, a, false, b3, (short)0, acc[3], false, false);
    }

    // C/D layout: VGPR v, lanes 0-15 -> M=v, lanes 16-31 -> M=v+8, N = lane&15
    const int mrow = mbase + (half << 3);
    float* __restrict__ Yb = Y + (size_t)mrow * LDY;

    if (mbase + 16 <= Co) {            // uniform full-tile fast path
        float bv[8];
#pragma unroll
        for (int v = 0; v < 8; ++v) bv[v] = bias ? bias[mrow + v] : 0.0f;
#pragma unroll
        for (int t = 0; t < 4; ++t) {
            int col = cbase + (t << 4) + l15;
#pragma unroll
            for (int v = 0; v < 8; ++v) {
                float val = acc[t][v];
                float* yp = Yb + (size_t)v * LDY + col;   // imm offset v*32KB
                if (mode == 0)      *yp  = val + bv[v];
                else if (mode == 1) *yp += val + bv[v];
                else                *yp += val;
            }
        }
    } else {                           // ragged last M tile
        float bv[8];
#pragma unroll
        for (int v = 0; v < 8; ++v) {
            int mo  = mrow + v;
            int moc = (mo < Co) ? mo : (Co - 1);
            float bt = bias ? bias[moc] : 0.0f;           // clamped, unconditional
            bv[v] = (mo < Co) ? bt : 0.0f;
        }
#pragma unroll
        for (int t = 0; t < 4; ++t) {
            int col = cbase + (t << 4) + l15;
#pragma unroll
            for (int v = 0; v < 8; ++v) {
                if (mrow + v < Co) {
                    float val = acc[t][v];
                    float* yp = Yb + (size_t)v * LDY + col;
                    if (mode == 0)      *yp  = val + bv[v];
                    else if (mode == 1) *yp += val + bv[v];
                    else                *yp += val;
                }
            }
        }
    }
}

// Pre-pack [Co,Ci,2,3] conv weights into dense [tap][Co*Ci] (tap = ky*3+kx).
__global__ void pack_wtaps(const float* __restrict__ Wsrc, float* __restrict__ Wd,
                           int CoCi, int ntot)
{
    int i = blockIdx.x * blockDim.x + threadIdx.x;
    if (i >= ntot) return;
    int tap = i / CoCi;
    int rem = i - tap * CoCi;          // = o*Ci + ci
    Wd[i] = Wsrc[rem * 6 + tap];
}

// Decode 8-bit binary planes -> img in [-1,1]; build hx (=img) and vx (shift
// down one row with learned top row).  img[n,c,w,h] = sum_b x[n,c,b,h,w]*2^(7-b)
__global__ void decode_k(const float* __restrict__ x,
                         const float* __restrict__ x2vx_pad,
                         float* __restrict__ HX, float* __restrict__ VX)
{
    int i = blockIdx.x * blockDim.x + threadIdx.x;
    if (i >= 3 * P_TOT) return;
    int c   = i >> 13;
    int rem = i & 8191;
    int n   = rem >> 10;
    int p   = rem & 1023;
    int w   = p >> 5;          // row (original W axis)
    int h   = p & 31;          // col (original H axis)
    float s = 0.f;
#pragma unroll
    for (int b = 0; b < 8; ++b)
        s += x[(((n * 3 + c) * 8 + b) * 32 + h) * 32 + w] * (float)(1 << (7 - b));
    float val = s * (1.0f / 127.5f) - 1.0f;
    HX[(size_t)c * P_TOT + n * 1024 + p] = val;
    if (w < 31) VX[(size_t)c * P_TOT + n * 1024 + (w + 1) * 32 + h] = val;
    if (w == 0) VX[(size_t)c * P_TOT + n * 1024 + h] = x2vx_pad[c * 32 + h];
}

// Build v_conv padded tensor: [Ci, N, H+d, W+2d] with learned px1/px2/py1.
__global__ void pack_vpad(const float* __restrict__ X,
                          const float* __restrict__ px1, const float* __restrict__ px2,
                          const float* __restrict__ py1,
                          float* __restrict__ XP, int Ci, int d, int ntot)
{
    int i = blockIdx.x * blockDim.x + threadIdx.x;
    if (i >= ntot) return;
    int HP = 32 + d, WP = 32 + 2 * d;
    int pc = i % WP; int t = i / WP;
    int pr = t % HP; t /= HP;
    int n  = t % 8;  int ci = t / 8;
    float v;
    if (pr < d) v = py1[(ci * d + pr) * WP + pc];
    else {
        int row = pr - d;
        if (pc < d)            v = px1[(ci * 32 + row) * d + pc];
        else if (pc >= 32 + d) v = px2[(ci * 32 + row) * d + (pc - (32 + d))];
        else                   v = X[(size_t)ci * P_TOT + n * 1024 + row * 32 + (pc - d)];
    }
    XP[i] = v;
}

// Masked-conv horizontal shift: XS = [pad(d cols) | X[..., :-d]]
__global__ void pack_xs(const float* __restrict__ X, const float* __restrict__ pad,
                        float* __restrict__ XS, int Ci, int d, int ntot)
{
    int i = blockIdx.x * blockDim.x + threadIdx.x;
    if (i >= ntot) return;
    int ci = i >> 13;
    int p  = i & 8191;
    int n = p >> 10, r = (p >> 5) & 31, cl = p & 31;
    XS[i] = (cl < d) ? pad[(ci * 32 + r) * d + cl]
                     : X[(size_t)ci * P_TOT + n * 1024 + r * 32 + (cl - d)];
}

// concat_minmax: channel pairs (2q,2q+1) -> (max, sum-max) interleaved.
__global__ void minmax_k(const float* __restrict__ X, float* __restrict__ Y, int ntot)
{
    int i = blockIdx.x * blockDim.x + threadIdx.x;
    if (i >= ntot) return;
    int q = i >> 13;
    int p = i & 8191;
    size_t i0 = (size_t)(2 * q) * P_TOT + p;
    float a = X[i0], b = X[i0 + P_TOT];
    float mx = fmaxf(a, b);
    Y[i0]          = mx;
    Y[i0 + P_TOT]  = a + b - mx;   // == min, matching reference arithmetic
}

// Per-channel batch stats (biased var) -> fused scale/shift. One WG per channel.
__global__ void bn_stats(const float* __restrict__ X,
                         const float* __restrict__ g, const float* __restrict__ b,
                         float* __restrict__ ss)
{
    int c = blockIdx.x;
    const float* xc = X + (size_t)c * P_TOT;
    float s = 0.f, s2 = 0.f;
    for (int i = threadIdx.x; i < P_TOT; i += 256) {
        float v = xc[i]; s += v; s2 += v * v;
    }
    __shared__ float sh[256], sh2[256];
    sh[threadIdx.x] = s; sh2[threadIdx.x] = s2;
    __syncthreads();
    for (int st = 128; st > 0; st >>= 1) {
        if (threadIdx.x < st) { sh[threadIdx.x] += sh[threadIdx.x + st];
                                sh2[threadIdx.x] += sh2[threadIdx.x + st]; }
        __syncthreads();
    }
    if (threadIdx.x == 0) {
        float mean = sh[0] * (1.0f / P_TOT);
        float var  = sh2[0] * (1.0f / P_TOT) - mean * mean;
        float rstd = rsqrtf(var + 1e-5f);
        float sc   = g[c] * rstd;
        ss[2 * c]     = sc;
        ss[2 * c + 1] = b[c] - mean * sc;
    }
}

// y = (res ? res : 0) + x*scale + shift
__global__ void bn_apply(const float* __restrict__ X, const float* __restrict__ ss,
                         const float* __restrict__ res, float* __restrict__ Y, int ntot)
{
    int i = blockIdx.x * blockDim.x + threadIdx.x;
    if (i >= ntot) return;
    int c = i >> 13;
    float v = X[i] * ss[2 * c] + ss[2 * c + 1];
    Y[i] = (res ? res[i] : 0.0f) + v;
}

__global__ void axpy_copy(const float* __restrict__ X, float* __restrict__ Y,
                          int ntot, int add)
{
    int i = blockIdx.x * blockDim.x + threadIdx.x;
    if (i >= ntot) return;
    Y[i] = add ? (Y[i] + X[i]) : X[i];
}

__global__ void fill_zero(float* __restrict__ Y, int ntot)
{
    int i = blockIdx.x * blockDim.x + threadIdx.x;
    if (i < ntot) Y[i] = 0.0f;
}

// O[24, 8192] channel-major -> d_out [N,3,8,32,32]
__global__ void store_out(const float* __restrict__ O, float* __restrict__ out)
{
    int i = blockIdx.x * blockDim.x + threadIdx.x;
    if (i >= 24 * P_TOT) return;
    int ch = i >> 13;
    int p  = i & 8191;
    int n  = p >> 10;
    int sp = p & 1023;
    out[(size_t)(n * 24 + ch) * 1024 + sp] = O[i];
}

// ---------------------------------------------------------------------------

extern "C" void kernel_launch(void* const* d_in, const int* in_sizes, int n_in,
                              void* d_out, int out_size, void* d_ws, size_t ws_size,
                              hipStream_t stream)
{
    (void)in_sizes; (void)n_in; (void)out_size; (void)ws_size;

    int ii = 0;
    auto nxt = [&]() -> const float* { return (const float*)d_in[ii++]; };

    const float* xbits = nxt();   // [8,3,8,32,32] f32 (0/1)

    struct Masked { const float *Wc,*Wr,*Wrg,*Wrgb,*bc,*br,*brg,*brgb,*pad; };
    struct Layer {
        Masked h2h;  const float *h_bn_b,*h_bn_g;
        Masked h_conv;
        const float *v2h_W,*v2h_b,*v_bn_b,*v_bn_g;
        const float *vW,*vb,*px1,*px2,*py1;
    } L[15];

    // Pytree flatten order: dict keys sorted ('W*' < 'b*' < 'p*'); list in order.
    for (int l = 0; l < 15; ++l) {
        bool A = (l == 0);
        Masked hh; hh.Wc = nullptr; hh.bc = nullptr; hh.pad = nullptr;
        hh.Wr = nxt(); hh.Wrg = nxt(); hh.Wrgb = nxt();
        hh.br = nxt(); hh.brg = nxt(); hh.brgb = nxt();
        L[l].h2h = hh;
        L[l].h_bn_b = nxt(); L[l].h_bn_g = nxt();
        Masked hc;
        hc.Wc = nxt(); hc.Wr = nxt(); hc.Wrg = nxt();
        hc.Wrgb = A ? nullptr : nxt();
        hc.bc = nxt(); hc.br = nxt(); hc.brg = nxt();
        hc.brgb = A ? nullptr : nxt();
        hc.pad = nxt();
        L[l].h_conv = hc;
        L[l].v2h_W = nxt(); L[l].v2h_b = nxt();
        L[l].v_bn_b = nxt(); L[l].v_bn_g = nxt();
        L[l].vW = nxt(); L[l].vb = nxt();
        L[l].px1 = nxt(); L[l].px2 = nxt(); L[l].py1 = nxt();
    }
    auto rdM = [&](Masked& M) {
        M.Wc = nxt(); M.Wr = nxt(); M.Wrg = nxt(); M.Wrgb = nxt();
        M.bc = nxt(); M.br = nxt(); M.brg = nxt(); M.brgb = nxt(); M.pad = nxt();
    };
    Masked O1, O2; rdM(O1); rdM(O2);
    const float* vx2vx_W  = nxt();
    const float* vx2vx_b  = nxt();
    const float* x2vx_pad = nxt();

    // Workspace carve (all f32). Largest buffer (33 MB) fits in the 192MB L2.
    float* w = (float*)d_ws;
    auto take = [&](size_t n) { float* p = w; w += n; return p; };
    const size_t CP = (size_t)192 * P_TOT;
    float *VX = take(CP), *VXN = take(CP), *HX = take(CP), *HXN = take(CP);
    float *VCONV = take(CP), *MM = take(CP), *HCONV = take(CP), *HACT = take(CP);
    float *SKIP = take(CP), *RES0 = take(CP);
    float *XPAD = take((size_t)192 * 8 * 36 * 40);
    float *WTAP = take((size_t)6 * 192 * 192);
    float *XS   = take((size_t)1020 * P_TOT);
    float *BIG1 = take((size_t)1020 * P_TOT);
    float *BIG2 = take((size_t)1020 * P_TOT);
    float *O24  = take((size_t)24 * P_TOT);
    float *SS   = take(2048);

    auto gemm = [&](const float* Wm, const float* bias,
                    const float* X, int chan_stride, int img_stride, int row_stride,
                    int sy, int sx, float* Y, int Co, int Ci, int mode) {
        dim3 g(P_TOT / 64, (unsigned)((Co + 15) / 16));
        wmma_gemm<<<g, dim3(32), 0, stream>>>(Wm, bias, X, chan_stride,
                                              img_stride, row_stride, sy, sx,
                                              Y, Co, Ci, mode);
    };
    auto gemm1x1 = [&](const float* Wm, const float* bias, const float* X,
                       float* Y, int Co, int Ci, int mode) {
        gemm(Wm, bias, X, P_TOT, 1024, 32, 0, 0, Y, Co, Ci, mode);
    };
    auto ew = [&](int n) { return dim3((unsigned)((n + 255) / 256)); };

    decode_k<<<ew(3 * P_TOT), 256, 0, stream>>>(xbits, x2vx_pad, HX, VX);

    static const int DIL[15] = {1,1,1,1,1, 2,2,2,2,2, 4,4,4,4,4};

    for (int l = 0; l < 15; ++l) {
        const int d  = DIL[l];
        const int Ci = (l == 0) ? 3 : 192;
        const int HP = 32 + d, WP = 32 + 2 * d;
        const int pimg = HP * WP, pchan = 8 * pimg;

        // --- vertical stack: pack taps + pad, 6-tap WMMA conv, minmax, BN ---
        pack_wtaps<<<ew(6 * 192 * Ci), 256, 0, stream>>>(L[l].vW, WTAP, 192 * Ci,
                                                         6 * 192 * Ci);
        pack_vpad<<<ew(Ci * pchan), 256, 0, stream>>>(VX, L[l].px1, L[l].px2,
                                                      L[l].py1, XPAD, Ci, d, Ci * pchan);
        for (int ky = 0; ky < 2; ++ky)
            for (int kx = 0; kx < 3; ++kx) {
                int tap  = ky * 3 + kx;
                int mode = (tap == 0) ? 0 : 2;
                gemm(WTAP + (size_t)tap * 192 * Ci, L[l].vb, XPAD, pchan, pimg, WP,
                     ky * d, kx * d, VCONV, 192, Ci, mode);
            }
        minmax_k<<<ew(96 * P_TOT), 256, 0, stream>>>(VCONV, MM, 96 * P_TOT);
        bn_stats<<<192, 256, 0, stream>>>(MM, L[l].v_bn_g, L[l].v_bn_b, SS);
        const float* resv;
        if (l == 0) { gemm1x1(vx2vx_W, vx2vx_b, VX, RES0, 192, 3, 0); resv = RES0; }
        else        { resv = VX; }
        bn_apply<<<ew(192 * P_TOT), 256, 0, stream>>>(MM, SS, resv, VXN, 192 * P_TOT);
        { float* t = VX; VX = VXN; VXN = t; }

        // --- horizontal stack: masked conv (range GEMMs + shifted Wc) ---
        const Masked& hc = L[l].h_conv;
        const int in_rg = Ci / 3;
        if (l == 0) {  // mask 'A': first 64 channels zero
            fill_zero<<<ew(64 * P_TOT), 256, 0, stream>>>(HCONV, 64 * P_TOT);
            gemm1x1(hc.Wr,  hc.br,  HX, HCONV + (size_t)64  * P_TOT, 64, in_rg,     0);
            gemm1x1(hc.Wrg, hc.brg, HX, HCONV + (size_t)128 * P_TOT, 64, 2 * in_rg, 0);
        } else {       // mask 'B'
            gemm1x1(hc.Wr,   hc.br,   HX, HCONV,                      64, in_rg,     0);
            gemm1x1(hc.Wrg,  hc.brg,  HX, HCONV + (size_t)64  * P_TOT, 64, 2 * in_rg, 0);
            gemm1x1(hc.Wrgb, hc.brgb, HX, HCONV + (size_t)128 * P_TOT, 64, 3 * in_rg, 0);
        }
        pack_xs<<<ew(Ci * P_TOT), 256, 0, stream>>>(HX, hc.pad, XS, Ci, d, Ci * P_TOT);
        gemm1x1(hc.Wc, hc.bc, XS, HCONV, 192, Ci, 1);
        gemm1x1(L[l].v2h_W, L[l].v2h_b, VCONV, HCONV, 192, 192, 1);

        minmax_k<<<ew(96 * P_TOT), 256, 0, stream>>>(HCONV, MM, 96 * P_TOT);
        bn_stats<<<192, 256, 0, stream>>>(MM, L[l].h_bn_g, L[l].h_bn_b, SS);
        bn_apply<<<ew(192 * P_TOT), 256, 0, stream>>>(MM, SS, (const float*)nullptr,
                                                      HACT, 192 * P_TOT);

        axpy_copy<<<ew(192 * P_TOT), 256, 0, stream>>>(HACT, SKIP, 192 * P_TOT,
                                                       l == 0 ? 0 : 1);

        // hx = res_hx + h2h(h_act)   (mask 'B', k=1)
        const Masked& hb = L[l].h2h;
        int hmode = 0;
        if (l > 0) {
            axpy_copy<<<ew(192 * P_TOT), 256, 0, stream>>>(HX, HXN, 192 * P_TOT, 0);
            hmode = 1;
        }
        gemm1x1(hb.Wr,   hb.br,   HACT, HXN,                       64, 64,  hmode);
        gemm1x1(hb.Wrg,  hb.brg,  HACT, HXN + (size_t)64  * P_TOT, 64, 128, hmode);
        gemm1x1(hb.Wrgb, hb.brgb, HACT, HXN + (size_t)128 * P_TOT, 64, 192, hmode);
        { float* t = HX; HX = HXN; HXN = t; }
    }

    // --- output head ---
    minmax_k<<<ew(96 * P_TOT), 256, 0, stream>>>(SKIP, MM, 96 * P_TOT);
    gemm1x1(O1.Wr,   O1.br,   MM, BIG1,                       340, 64,  0);
    gemm1x1(O1.Wrg,  O1.brg,  MM, BIG1 + (size_t)340 * P_TOT, 340, 128, 0);
    gemm1x1(O1.Wrgb, O1.brgb, MM, BIG1 + (size_t)680 * P_TOT, 340, 192, 0);
    pack_xs<<<ew(192 * P_TOT), 256, 0, stream>>>(MM, O1.pad, XS, 192, 1, 192 * P_TOT);
    gemm1x1(O1.Wc, O1.bc, XS, BIG1, 1020, 192, 1);

    minmax_k<<<ew(510 * P_TOT), 256, 0, stream>>>(BIG1, BIG2, 510 * P_TOT);

    gemm1x1(O2.Wr,   O2.br,   BIG2, O24,                      8, 340,  0);
    gemm1x1(O2.Wrg,  O2.brg,  BIG2, O24 + (size_t)8  * P_TOT, 8, 680,  0);
    gemm1x1(O2.Wrgb, O2.brgb, BIG2, O24 + (size_t)16 * P_TOT, 8, 1020, 0);
    pack_xs<<<ew(1020 * P_TOT), 256, 0, stream>>>(BIG2, O2.pad, XS, 1020, 1, 1020 * P_TOT);
    gemm1x1(O2.Wc, O2.bc, XS, O24, 24, 1020, 1);

    store_out<<<ew(24 * P_TOT), 256, 0, stream>>>(O24, (float*)d_out);
}